// RX_50259707298061
// MI455X (gfx1250) — compile-verified
//
#include <hip/hip_runtime.h>

// RX gate on wire 0 of a 24-qubit state vector.
// Memory-bound streaming butterfly (0.25 FLOP/byte): optimal CDNA5 shape is
// b128 non-temporal global loads/stores under wave32. Trig is hoisted into a
// 1-thread setup kernel so the hot kernel carries no libm argument-reduction
// code (it was 350+ SALU instructions of Payne-Hanek in-line).

typedef float v4f __attribute__((ext_vector_type(4)));

#define WIRES  24
#define NSTATE (1 << WIRES)   // 16777216 amplitudes
#define HALF   (NSTATE >> 1)  // 8388608: stride between wire-0 partners

// --- setup: compute {cos(theta/2), sin(theta/2)} once, accurately ---------
__global__ void rx_angle_setup(const float* __restrict__ angle,
                               float* __restrict__ cs) {
    const float ht = 0.5f * angle[0];
    cs[0] = cosf(ht);   // accurate libm path, executed by exactly one thread
    cs[1] = sinf(ht);
}

// --- hot kernel: pure streaming butterfly --------------------------------
__global__ __launch_bounds__(256) void rx_wire0_kernel(
    const float* __restrict__ xr,
    const float* __restrict__ xi,
    const float* __restrict__ cs,   // {c, s} in workspace
    float* __restrict__ out)
{
    // Uniform address -> s_load_b64, broadcast to all lanes.
    const float c = cs[0];
    const float s = cs[1];

    const size_t i = ((size_t)blockIdx.x * blockDim.x + threadIdx.x) * 4;

    // 4 x global_load_b128 th:NT — every byte read exactly once; 256 MB
    // working set exceeds the 192 MB L2, so don't cache.
    const v4f xr0 = __builtin_nontemporal_load((const v4f*)(xr + i));
    const v4f xr1 = __builtin_nontemporal_load((const v4f*)(xr + i + HALF));
    const v4f xi0 = __builtin_nontemporal_load((const v4f*)(xi + i));
    const v4f xi1 = __builtin_nontemporal_load((const v4f*)(xi + i + HALF));

    // out0 = c*x0 - i*s*x1 ; out1 = -i*s*x0 + c*x1   (complex, c,s real)
    const v4f or0 = c * xr0 + s * xi1;
    const v4f or1 = c * xr1 + s * xi0;
    const v4f oi0 = c * xi0 - s * xr1;
    const v4f oi1 = c * xi1 - s * xr0;

    float* __restrict__ out_r = out;           // d_out[0, :]
    float* __restrict__ out_i = out + NSTATE;  // d_out[1, :]

    // 4 x global_store_b128 th:NT — streamed output, never re-read.
    __builtin_nontemporal_store(or0, (v4f*)(out_r + i));
    __builtin_nontemporal_store(or1, (v4f*)(out_r + i + HALF));
    __builtin_nontemporal_store(oi0, (v4f*)(out_i + i));
    __builtin_nontemporal_store(oi1, (v4f*)(out_i + i + HALF));
}

extern "C" void kernel_launch(void* const* d_in, const int* in_sizes, int n_in,
                              void* d_out, int out_size, void* d_ws, size_t ws_size,
                              hipStream_t stream) {
    const float* xr    = (const float*)d_in[0];  // x_real, N floats
    const float* xi    = (const float*)d_in[1];  // x_imag, N floats
    const float* angle = (const float*)d_in[2];  // 24 floats, only [0] used
    float* out = (float*)d_out;                  // [2, N] float32
    float* cs  = (float*)d_ws;                   // 2 floats of scratch

    rx_angle_setup<<<1, 1, 0, stream>>>(angle, cs);

    constexpr int threads = 256;                           // 8 wave32 waves/block
    constexpr int elems_per_thread = 4;                    // one float4 per half
    constexpr int total_threads = HALF / elems_per_thread; // 2,097,152
    constexpr int blocks = total_threads / threads;        // 8,192

    rx_wire0_kernel<<<blocks, threads, 0, stream>>>(xr, xi, cs, out);
}